// DotProductAttention_300647710747
// MI455X (gfx1250) — compile-verified
//
#include <hip/hip_runtime.h>

// Problem constants (fixed by the reference):
#define NKD   8
#define NVD   8
#define CD    80            // NK + NK*NV + NV
#define HWD   (512 * 512)   // spatial size per (b, c) plane
#define BD    4
#define TILE  128           // pixels per tile (divides HWD; 2048 tiles per batch)
#define NT    8             // tiles per workgroup (divides 2048 -> no batch cross)
#define BLOCK 64            // threads per workgroup (2 wave32)
#define PPT   2             // pixels per thread (TILE / BLOCK)

typedef float        f32x2 __attribute__((ext_vector_type(2)));
typedef unsigned int u32x4 __attribute__((ext_vector_type(4)));
typedef unsigned int u32x8 __attribute__((ext_vector_type(8)));

__global__ __launch_bounds__(BLOCK)
void DotProductAttention_gfx1250_kernel(const float* __restrict__ inp,
                                        float* __restrict__ out)
{
    // Double-buffered channel-major input staging: buf[stage][channel][pixel].
    // TDM writes the 80 strided channel rows back-to-back -> exactly this layout.
    __shared__ float buf[2][CD][TILE];    // 80 KB
    // Double-buffered output staging, drained by tensor_store_from_lds.
    __shared__ float obuf[2][NVD][TILE];  //  8 KB

    const int  t      = threadIdx.x;      // 0..63
    const bool leader = (t < 32);         // wave 0 drives the Tensor Data Mover
    const long tile0  = (long)blockIdx.x * NT;

    // ---- Loop-invariant TDM descriptor material (uniform -> SGPRs) ----
    // All NT tiles of this WG live in one batch: hoist batch + base addresses.
    const long pix00 = tile0 * (long)TILE;
    const int  b     = (int)(pix00 / HWD);
    const int  p00   = (int)(pix00 % HWD);
    const unsigned long long gaL0 =
        (unsigned long long)(size_t)(inp + (size_t)b * CD * HWD + p00);
    const unsigned long long gaS0 =
        (unsigned long long)(size_t)(out + (size_t)b * NVD * HWD + p00);
    const unsigned ldsL[2] = { (unsigned)(size_t)(&buf[0][0][0]),
                               (unsigned)(size_t)(&buf[1][0][0]) };
    const unsigned ldsS[2] = { (unsigned)(size_t)(&obuf[0][0][0]),
                               (unsigned)(size_t)(&obuf[1][0][0]) };

    // Constant D# group1 (2D tile): dim0 = TILE contiguous floats (pixels),
    // dim1 = `rows` channel rows with stride HWD elements, data_size = 4 B.
    auto mk_g1 = [](unsigned rows) -> u32x8 {
        u32x8 g1;
        g1[0] = (2u << 16);             // workgroup_mask=0, data_size=2 (4 bytes)
        g1[1] = ((unsigned)TILE) << 16; // atomic_barrier_addr=0 | tensor_dim0[15:0]
        g1[2] = rows << 16;             // tensor_dim0[31:16]=0  | tensor_dim1[15:0]
        g1[3] = ((unsigned)TILE) << 16; // tensor_dim1[31:16]=0  | tile_dim0=TILE
        g1[4] = rows;                   // tile_dim1=rows | tile_dim2=0
        g1[5] = (unsigned)HWD;          // tensor_dim0_stride[31:0] = 262144 elems
        g1[6] = 0u;                     // stride[47:32]=0 | tensor_dim1_stride lo=0
        g1[7] = 0u;                     // tensor_dim1_stride hi = 0 (2D tile)
        return g1;
    };
    const u32x8 g1L = mk_g1((unsigned)CD);   // input: 80 rows
    const u32x8 g1S = mk_g1((unsigned)NVD);  // output: 8 rows

    auto mk_g0 = [](unsigned ldsAddr, unsigned long long ga) -> u32x4 {
        u32x4 g0;
        g0[0] = 1u;                                    // count=1 (valid), user mode
        g0[1] = ldsAddr;                               // lds_addr (bytes)
        g0[2] = (unsigned)ga;                          // global_addr[31:0]
        g0[3] = (unsigned)((ga >> 32) & 0x1ffffffull)  // global_addr[56:32]
              | (2u << 30);                            // type=2 ("image")
        return g0;
    };

    auto stageL = [&](int s, int it) {       // global -> LDS (input tile)
        const u32x4 g0 = mk_g0(ldsL[s], gaL0 + (unsigned long long)it * (TILE * 4));
        asm volatile("tensor_load_to_lds %0, %1" :: "s"(g0), "s"(g1L) : "memory");
    };
    auto stageS = [&](int s, int it) {       // LDS -> global (output tile)
        const u32x4 g0 = mk_g0(ldsS[s], gaS0 + (unsigned long long)it * (TILE * 4));
        asm volatile("tensor_store_from_lds %0, %1" :: "s"(g0), "s"(g1S) : "memory");
    };

    if (leader) { stageL(0, 0); stageL(1, 1); }

    for (int it = 0; it < NT; ++it) {
        const int s = it & 1;

        // TENSORcnt completes in issue order. Ops younger than L_it are
        // {S_{it-1}, L_{it+1}} when both exist -> wait <=2 proves L_it landed
        // AND S_{it-2} (3+ ops older) finished, so obuf[s] is reusable.
        if (leader) {
            if (it == 0)           __builtin_amdgcn_s_wait_tensorcnt(1);
            else if (it + 1 < NT)  __builtin_amdgcn_s_wait_tensorcnt(2);
            else                   __builtin_amdgcn_s_wait_tensorcnt(1);
        }
        __syncthreads();   // DMA-written LDS visible; obuf[s] free for reuse

        const int col = t * PPT;                        // even -> 8B aligned

        // q[8] for both pixels (b64 LDS reads, conflict-free bank sweep)
        f32x2 qv[NKD];
#pragma unroll
        for (int kk = 0; kk < NKD; ++kk)
            qv[kk] = *(const f32x2*)&buf[s][kk][col];

        // qk[v] = sum_k q[k] * K[k][v]   (channel 8 + k*8 + v)
        f32x2 acc[NVD];
#pragma unroll
        for (int vv = 0; vv < NVD; ++vv) acc[vv] = (f32x2){0.0f, 0.0f};
#pragma unroll
        for (int kk = 0; kk < NKD; ++kk) {
            const f32x2 qk = qv[kk];
#pragma unroll
            for (int vv = 0; vv < NVD; ++vv)
                acc[vv] += qk * (*(const f32x2*)&buf[s][NKD + kk * NVD + vv][col]);
        }

        // softmax over the 8 v-channels (scaled by 1/sqrt(8)), per pixel
        const float scale = 0.35355339059327373f;
        f32x2 w[NVD];
#pragma unroll
        for (int h = 0; h < PPT; ++h) {
            float z[NVD];
#pragma unroll
            for (int vv = 0; vv < NVD; ++vv) z[vv] = acc[vv][h] * scale;
            float m = z[0];
#pragma unroll
            for (int vv = 1; vv < NVD; ++vv) m = fmaxf(m, z[vv]);
            float e[NVD];
            float ssum = 0.0f;
#pragma unroll
            for (int vv = 0; vv < NVD; ++vv) { e[vv] = __expf(z[vv] - m); ssum += e[vv]; }
#if __has_builtin(__builtin_amdgcn_rcpf)
            const float inv = __builtin_amdgcn_rcpf(ssum);
#else
            const float inv = 1.0f / ssum;
#endif
#pragma unroll
            for (int vv = 0; vv < NVD; ++vv) w[vv][h] = e[vv] * inv;
        }

        // attn[v] * v[v] -> output staging tile (b64 DS writes, conflict-free)
#pragma unroll
        for (int vv = 0; vv < NVD; ++vv) {
            const f32x2 val = *(const f32x2*)&buf[s][NKD + NKD * NVD + vv][col];
            *(f32x2*)&obuf[s][vv][col] = w[vv] * val;
        }

        // Barrier drains every wave's DScnt: obuf[s] complete, buf[s] fully read.
        __syncthreads();
        if (leader) {
            stageS(s, it);                       // DMA results out
            if (it + 2 < NT) stageL(s, it + 2);  // refill freed input buffer
        }
    }
    // S_ENDPGM performs an implicit wait-idle, draining wave 0's TENSORcnt.
}

extern "C" void kernel_launch(void* const* d_in, const int* in_sizes, int n_in,
                              void* d_out, int out_size, void* d_ws, size_t ws_size,
                              hipStream_t stream)
{
    (void)in_sizes; (void)n_in; (void)d_ws; (void)ws_size; (void)out_size;
    const float* inp = (const float*)d_in[0];
    float* out = (float*)d_out;

    const long totalPix = (long)BD * HWD;          // 1,048,576
    const int  tiles    = (int)(totalPix / TILE);  // 8,192
    const int  wgs      = tiles / NT;              // 1,024 workgroups x 2 waves

    DotProductAttention_gfx1250_kernel<<<wgs, BLOCK, 0, stream>>>(inp, out);
}